// AATPE_82978768159404
// MI455X (gfx1250) — compile-verified
//
#include <hip/hip_runtime.h>
#include <hip/hip_bf16.h>
#include <stdint.h>

// ---------------------------------------------------------------------------
// Blocked-sparse  y = eps*diag*x + A (diag*mask*(A^T x)) * mask,  D=4 blocks.
// Bandwidth-bound: boo_values streamed twice (~410 MB of ~465 MB total) ->
// ~20 us floor at 23.3 TB/s. WMMA is inapplicable (per-edge block-diagonal
// 4x4 matvecs). CDNA5 path: TDM tensor_load_to_lds streams 16 KB tiles of
// boo_values into LDS with NT hint (don't thrash L2's hot 5 MB working set),
// LDS-padded to an 80 B per-edge stride (16 B aligned -> legal ds_load_b128),
// double-buffered with s_wait_tensorcnt + workgroup barriers.
// ---------------------------------------------------------------------------

#define EPS      0.01f
#define TILE_E   256                 // edges per TDM tile (1 per thread)
#define EB_DW    20                  // dwords per edge block in LDS (16 + 4 pad)
#define TILE_SRC_DW (TILE_E * 16)    // 4096 dwords = 16 KB from global
#define TILE_LDS_DW (TILE_E * EB_DW) // 5120 dwords = 20 KB in LDS

typedef __attribute__((ext_vector_type(4))) unsigned int v4u;
typedef __attribute__((ext_vector_type(8))) int          v8i;
typedef __attribute__((ext_vector_type(4))) int          v4i;

__device__ __forceinline__ uint32_t lds_byte_addr(const void* p) {
    // low 32 bits of a flat shared-aperture address are the LDS byte offset
    return (uint32_t)(uintptr_t)p;
}

// One TDM descriptor = one contiguous 16 KB tile of boo_values -> LDS,
// padded 4 dwords per 16 dwords on the LDS side (80 B edge stride).
__device__ __forceinline__ void tdm_load_tile(const float* boo, int tile,
                                              uint32_t lds_addr) {
    uint64_t ga = (uint64_t)(uintptr_t)boo + (uint64_t)tile * (TILE_E * 64ull);

    v4u g0;
    g0.x = 1u;                                   // count=1 (valid user D#)
    g0.y = lds_addr;                             // lds_addr (bytes)
    g0.z = (uint32_t)ga;                         // global_addr[31:0]
    g0.w = (uint32_t)((ga >> 32) & 0x1FFFFFFu)   // global_addr[56:32]
         | (2u << 30);                           // type = 2 ("image")

    v8i g1;
    // wg_mask=0 | data_size=4B(2)<<16 | pad_en<<20 | pad_interval=16dw(3)<<22
    // | pad_amount=4dw(3)<<25
    g1[0] = (int)((2u << 16) | (1u << 20) | (3u << 22) | (3u << 25));
    g1[1] = (int)(((uint32_t)TILE_SRC_DW & 0xFFFFu) << 16); // tensor_dim0 lo
    g1[2] = (int)((((uint32_t)TILE_SRC_DW >> 16) & 0xFFFFu) // tensor_dim0 hi
                  | (1u << 16));                            // tensor_dim1 = 1
    g1[3] = (int)(((uint32_t)TILE_SRC_DW & 0xFFFFu) << 16); // tile_dim0
    g1[4] = 1;                                              // tile_dim1=1
    g1[5] = (int)(uint32_t)TILE_SRC_DW;                     // dim0_stride lo
    g1[6] = 0;
    g1[7] = 0;

    v4i z4 = {0, 0, 0, 0};
    v8i z8 = {0, 0, 0, 0, 0, 0, 0, 0};
    __builtin_amdgcn_tensor_load_to_lds(g0, g1, z4, z4, z8, /*cpol=TH_NT*/ 1);
}

__global__ __launch_bounds__(256) void AATPE_init(float* acc0, float* acc1, int nd) {
    int i = blockIdx.x * blockDim.x + threadIdx.x;
    if (i < nd) { acc0[i] = 0.0f; acc1[i] = 0.0f; }
}

// Pass 1: acc0[col] += B^T x[row]
__global__ __launch_bounds__(256) void AATPE_pass1(
    const float* __restrict__ x, const int* __restrict__ ei,
    const float* __restrict__ boo, float* __restrict__ acc0,
    int E, int ntiles)
{
    __shared__ __align__(16) uint32_t smem[2][TILE_LDS_DW];
    const int  tid = (int)threadIdx.x;
    const bool w0  = (tid < 32);
    int tile = (int)blockIdx.x;
    if (tile >= ntiles) return;

    if (w0) {
        tdm_load_tile(boo, tile, lds_byte_addr(&smem[0][0]));
        __builtin_amdgcn_s_wait_tensorcnt(0);
    }
    __syncthreads();

    const int stride = (int)gridDim.x;
    int p = 0;
    for (; tile < ntiles; tile += stride) {
        const int nxt = tile + stride;
        if (w0 && nxt < ntiles)
            tdm_load_tile(boo, nxt, lds_byte_addr(&smem[p ^ 1][0]));

        const int e = tile * TILE_E + tid;
        if (e < E) {
            const int r = ei[e];
            const int c = ei[E + e];
            const float4 xr = ((const float4*)x)[r];
            const float4* eb = (const float4*)&smem[p][tid * EB_DW];
            const float4 r0 = eb[0];   // B row 0
            const float4 r1 = eb[1];   // B row 1
            const float4 r2 = eb[2];   // B row 2
            const float4 r3 = eb[3];   // B row 3
            // msg_t[i] = sum_j B[j][i] * x[j]  (column dot)
            float m0 = xr.x*r0.x + xr.y*r1.x + xr.z*r2.x + xr.w*r3.x;
            float m1 = xr.x*r0.y + xr.y*r1.y + xr.z*r2.y + xr.w*r3.y;
            float m2 = xr.x*r0.z + xr.y*r1.z + xr.z*r2.z + xr.w*r3.z;
            float m3 = xr.x*r0.w + xr.y*r1.w + xr.z*r2.w + xr.w*r3.w;
            atomicAdd(&acc0[c*4+0], m0);
            atomicAdd(&acc0[c*4+1], m1);
            atomicAdd(&acc0[c*4+2], m2);
            atomicAdd(&acc0[c*4+3], m3);
        }
        if (w0) __builtin_amdgcn_s_wait_tensorcnt(0);
        __syncthreads();
        p ^= 1;
    }
}

// Pass 2: acc1[row] += B * (acc0[col]*diag[col]*mask[col])
__global__ __launch_bounds__(256) void AATPE_pass2(
    const float* __restrict__ acc0, const int* __restrict__ ei,
    const float* __restrict__ boo, const float* __restrict__ diag,
    const float* __restrict__ mask, float* __restrict__ acc1,
    int E, int ntiles)
{
    __shared__ __align__(16) uint32_t smem[2][TILE_LDS_DW];
    const int  tid = (int)threadIdx.x;
    const bool w0  = (tid < 32);
    int tile = (int)blockIdx.x;
    if (tile >= ntiles) return;

    if (w0) {
        tdm_load_tile(boo, tile, lds_byte_addr(&smem[0][0]));
        __builtin_amdgcn_s_wait_tensorcnt(0);
    }
    __syncthreads();

    const int stride = (int)gridDim.x;
    int p = 0;
    for (; tile < ntiles; tile += stride) {
        const int nxt = tile + stride;
        if (w0 && nxt < ntiles)
            tdm_load_tile(boo, nxt, lds_byte_addr(&smem[p ^ 1][0]));

        const int e = tile * TILE_E + tid;
        if (e < E) {
            const int r = ei[e];
            const int c = ei[E + e];
            const float mk = mask[c];
            const float v0 = acc0[c*4+0] * diag[c*4+0] * mk;
            const float v1 = acc0[c*4+1] * diag[c*4+1] * mk;
            const float v2 = acc0[c*4+2] * diag[c*4+2] * mk;
            const float v3 = acc0[c*4+3] * diag[c*4+3] * mk;
            const float4* eb = (const float4*)&smem[p][tid * EB_DW];
            const float4 r0 = eb[0];
            const float4 r1 = eb[1];
            const float4 r2 = eb[2];
            const float4 r3 = eb[3];
            // msg[i] = sum_j B[i][j] * v[j]  (row dot)
            float m0 = v0*r0.x + v1*r0.y + v2*r0.z + v3*r0.w;
            float m1 = v0*r1.x + v1*r1.y + v2*r1.z + v3*r1.w;
            float m2 = v0*r2.x + v1*r2.y + v2*r2.z + v3*r2.w;
            float m3 = v0*r3.x + v1*r3.y + v2*r3.z + v3*r3.w;
            atomicAdd(&acc1[r*4+0], m0);
            atomicAdd(&acc1[r*4+1], m1);
            atomicAdd(&acc1[r*4+2], m2);
            atomicAdd(&acc1[r*4+3], m3);
        }
        if (w0) __builtin_amdgcn_s_wait_tensorcnt(0);
        __syncthreads();
        p ^= 1;
    }
}

__global__ __launch_bounds__(256) void AATPE_final(
    const float* __restrict__ x, const float* __restrict__ diag,
    const float* __restrict__ mask, const float* __restrict__ acc1,
    float* __restrict__ out, int nd)
{
    int i = blockIdx.x * blockDim.x + threadIdx.x;
    if (i < nd)
        out[i] = EPS * x[i] * diag[i] + mask[i >> 2] * acc1[i];
}

extern "C" void kernel_launch(void* const* d_in, const int* in_sizes, int n_in,
                              void* d_out, int out_size, void* d_ws, size_t ws_size,
                              hipStream_t stream)
{
    const float* x    = (const float*)d_in[0];
    const int*   ei   = (const int*)d_in[1];
    const float* boo  = (const float*)d_in[2];
    const float* mask = (const float*)d_in[3];
    const float* diag = (const float*)d_in[4];
    float* out = (float*)d_out;

    const int ND = in_sizes[0];        // N*4
    const int E  = in_sizes[1] / 2;

    float* acc0 = (float*)d_ws;        // [N,4] A^T x accumulator
    float* acc1 = acc0 + ND;           // [N,4] A (diag A^T x) accumulator

    const int ntiles = (E + TILE_E - 1) / TILE_E;
    int grid = ntiles < 2048 ? ntiles : 2048;

    AATPE_init <<<(ND + 255) / 256, 256, 0, stream>>>(acc0, acc1, ND);
    AATPE_pass1<<<grid,            256, 0, stream>>>(x,  ei, boo, acc0, E, ntiles);
    AATPE_pass2<<<grid,            256, 0, stream>>>(acc0, ei, boo, diag, mask, acc1, E, ntiles);
    AATPE_final<<<(ND + 255) / 256, 256, 0, stream>>>(x, diag, mask, acc1, out, ND);
}